// Classifier_55783035241052
// MI455X (gfx1250) — compile-verified
//
#include <hip/hip_runtime.h>
#include <math.h>

typedef __attribute__((ext_vector_type(2))) float v2f;
typedef __attribute__((ext_vector_type(8))) float v8f;

#define POOLB 49          // 7x7 bins (padded to 64 in pooledT)
#define POOLPAD 64
#define FEAT 2048
#define RED 512
#define FLATK 25088       // 512*49
#define H1N 4096
#define H2N 2048
#define NCLS 21
#define NREG 80

// ---------------------------------------------------------------------------
// Stage A: adaptive average pool per (roi, channel, bin).
// crop = base_x[0, c, x:x+w, y:y+h]  -> shape (C, Hc=w, Wc=h)
// Output TRANSPOSED + ZERO-PADDED for the WMMA B operand:
//   pooledT[r][bin(64)][c],  bins 49..63 = 0
// ---------------------------------------------------------------------------
__global__ void roi_pool_kernel(const float* __restrict__ base_x,
                                const int* __restrict__ rois,
                                float* __restrict__ pooledT) {
    const int c = blockIdx.x;          // 0..2047
    const int r = blockIdx.y;          // 0..3
    const int bin = threadIdx.x;       // 0..63 (49 real bins + 15 zero pads)

    float val = 0.f;
    if (bin < POOLB) {
        const int i = bin / 7, j = bin % 7;
        const int x = rois[r * 4 + 0];
        const int y = rois[r * 4 + 1];
        const int h = rois[r * 4 + 2];
        const int w = rois[r * 4 + 3];
        const int Hc = w, Wc = h;
        const int hs = (i * Hc) / 7;
        const int he = ((i + 1) * Hc + 6) / 7;
        const int ws_ = (j * Wc) / 7;
        const int we = ((j + 1) * Wc + 6) / 7;
        const float* plane = base_x + (size_t)c * 128 * 128;
        float s = 0.f;
        for (int a = hs; a < he; ++a) {
            const float* row = plane + (size_t)(x + a) * 128 + y;
            for (int b = ws_; b < we; ++b) s += row[b];
        }
        val = s / (float)((he - hs) * (we - ws_));
    }
    pooledT[((size_t)r * POOLPAD + bin) * FEAT + c] = val;
}

// ---------------------------------------------------------------------------
// Stage B: 1x1 conv as WMMA GEMM per roi: red_r = conv_w(512x2048) @ pooled_r(2048x49)
// One wave per 16(M) x 16(N) tile; K-loop of V_WMMA_F32_16X16X4_F32.
// f32 operand layouts per CDNA5 ISA 7.12.2 (lane L, kb = (L/16)*2):
//   A 16x4 : lane holds A[L%16][kb], A[L%16][kb+1]   -> b64 from conv_w row
//   B 4x16 : lane holds B[kb][L%16], B[kb+1][L%16]   -> b64 from pooledT row
//   D 16x16: vgpr v, lane L -> row v + 8*(L/16), col L%16
// Zero-padded pooledT makes all inner-loop loads unconditional.
// Output in flat layout: flat[r][o*49 + n]
// ---------------------------------------------------------------------------
__global__ void conv1x1_wmma_kernel(const float* __restrict__ conv_w,
                                    const float* __restrict__ conv_b,
                                    const float* __restrict__ pooledT,
                                    float* __restrict__ flat) {
    const int lane = threadIdx.x;             // 0..31 (one wave per block)
    const int m0 = blockIdx.x * 16;           // output-channel tile (512/16 = 32)
    const int n0 = blockIdx.y * 16;           // spatial tile (49 -> 64)
    const int r  = blockIdx.z;                // roi
    const int mrow = lane & 15;
    const int kb = (lane >> 4) * 2;           // 0 or 2
    const int n = n0 + mrow;

    const float* Arow = conv_w + (size_t)(m0 + mrow) * FEAT + kb;
    const float* Brow = pooledT + ((size_t)r * POOLPAD + n) * FEAT + kb;

    v8f c = {};
    for (int k = 0; k < FEAT; k += 4) {
        const v2f a = *(const v2f*)(Arow + k);   // unconditional global_load_b64
        const v2f b = *(const v2f*)(Brow + k);   // unconditional global_load_b64
        c = __builtin_amdgcn_wmma_f32_16x16x4_f32(
                false, a, false, b, (short)0, c, false, false);
    }

    if (n < POOLB) {
        const int half = lane >> 4;           // 0 -> rows 0..7, 1 -> rows 8..15
        #pragma unroll
        for (int v = 0; v < 8; ++v) {
            const int o = m0 + v + half * 8;
            flat[(size_t)r * FLATK + (size_t)o * POOLB + n] = c[v] + conv_b[o];
        }
    }
}

// ---------------------------------------------------------------------------
// Stages C/D/E: batch-4 GEMV, one wave per output row n.
// Y[r][n] = act( sum_k X[r][k] * W[n][k] + bias[n] ),  r = 0..3
// Lanes span K with float4 loads -> fully coalesced streaming of W.
// K must be a multiple of 128.
// ---------------------------------------------------------------------------
template <int K, bool RELU>
__global__ void gemv4_kernel(const float* __restrict__ W,
                             const float* __restrict__ bias,
                             const float* __restrict__ X,
                             float* __restrict__ Y, int N) {
    const int wave = (int)((blockIdx.x * blockDim.x + threadIdx.x) >> 5);
    const int lane = threadIdx.x & 31;
    if (wave >= N) return;
    const float* w = W + (size_t)wave * K;

    float a0 = 0.f, a1 = 0.f, a2 = 0.f, a3 = 0.f;
    for (int k = lane * 4; k < K; k += 128) {
        const float4 wv = *(const float4*)(w + k);
        const float4 x0 = *(const float4*)(X + 0 * K + k);
        const float4 x1 = *(const float4*)(X + 1 * K + k);
        const float4 x2 = *(const float4*)(X + 2 * K + k);
        const float4 x3 = *(const float4*)(X + 3 * K + k);
        a0 += wv.x * x0.x + wv.y * x0.y + wv.z * x0.z + wv.w * x0.w;
        a1 += wv.x * x1.x + wv.y * x1.y + wv.z * x1.z + wv.w * x1.w;
        a2 += wv.x * x2.x + wv.y * x2.y + wv.z * x2.z + wv.w * x2.w;
        a3 += wv.x * x3.x + wv.y * x3.y + wv.z * x3.z + wv.w * x3.w;
    }
    #pragma unroll
    for (int off = 16; off > 0; off >>= 1) {
        a0 += __shfl_xor(a0, off, 32);
        a1 += __shfl_xor(a1, off, 32);
        a2 += __shfl_xor(a2, off, 32);
        a3 += __shfl_xor(a3, off, 32);
    }
    if (lane == 0) {
        const float b = bias ? bias[wave] : 0.f;
        float r0 = a0 + b, r1 = a1 + b, r2 = a2 + b, r3 = a3 + b;
        if (RELU) {
            r0 = fmaxf(r0, 0.f); r1 = fmaxf(r1, 0.f);
            r2 = fmaxf(r2, 0.f); r3 = fmaxf(r3, 0.f);
        }
        Y[0 * (size_t)N + wave] = r0;
        Y[1 * (size_t)N + wave] = r1;
        Y[2 * (size_t)N + wave] = r2;
        Y[3 * (size_t)N + wave] = r3;
    }
}

// ---------------------------------------------------------------------------
// Softmax over 21 classes per roi (wave32 reductions).
// ---------------------------------------------------------------------------
__global__ void softmax21_kernel(const float* __restrict__ logits,
                                 float* __restrict__ out) {
    const int r = blockIdx.x;
    const int lane = threadIdx.x;
    float v = (lane < NCLS) ? logits[r * NCLS + lane] : -INFINITY;
    float m = v;
    #pragma unroll
    for (int off = 16; off > 0; off >>= 1) m = fmaxf(m, __shfl_xor(m, off, 32));
    float e = (lane < NCLS) ? expf(v - m) : 0.f;
    float s = e;
    #pragma unroll
    for (int off = 16; off > 0; off >>= 1) s += __shfl_xor(s, off, 32);
    if (lane < NCLS) out[r * NCLS + lane] = e / s;
}

extern "C" void kernel_launch(void* const* d_in, const int* in_sizes, int n_in,
                              void* d_out, int out_size, void* d_ws, size_t ws_size,
                              hipStream_t stream) {
    const float* base_x = (const float*)d_in[0];
    const int*   rois   = (const int*)d_in[1];
    const float* conv_w = (const float*)d_in[2];
    const float* conv_b = (const float*)d_in[3];
    const float* d1_w   = (const float*)d_in[4];
    const float* d1_b   = (const float*)d_in[5];
    const float* d2_w   = (const float*)d_in[6];
    const float* d2_b   = (const float*)d_in[7];
    const float* d3_w   = (const float*)d_in[8];
    const float* d4_w   = (const float*)d_in[9];
    float* out = (float*)d_out;

    float* ws      = (float*)d_ws;
    float* pooledT = ws;                              // 4*64*2048 = 524288
    float* flat    = pooledT + 4 * POOLPAD * FEAT;    // 4*25088   = 100352
    float* h1      = flat + 4 * FLATK;                // 4*4096    = 16384
    float* h2      = h1 + 4 * H1N;                    // 4*2048    = 8192
    float* cls     = h2 + 4 * H2N;                    // 4*21      = 84

    // A: adaptive ROI pooling (writes transposed + zero-padded pooledT)
    roi_pool_kernel<<<dim3(FEAT, 4), POOLPAD, 0, stream>>>(base_x, rois, pooledT);

    // B: 1x1 conv via f32 WMMA (M=512, N=49->64, K=2048 per roi)
    conv1x1_wmma_kernel<<<dim3(RED / 16, 4, 4), 32, 0, stream>>>(conv_w, conv_b,
                                                                 pooledT, flat);

    // C: h1 = relu(flat @ d1_w.T + d1_b)   -- 411 MB stream, coalesced GEMV
    gemv4_kernel<FLATK, true><<<(H1N + 7) / 8, 256, 0, stream>>>(d1_w, d1_b,
                                                                 flat, h1, H1N);
    // D: h2 = relu(h1 @ d2_w.T + d2_b)
    gemv4_kernel<H1N, true><<<(H2N + 7) / 8, 256, 0, stream>>>(d2_w, d2_b,
                                                               h1, h2, H2N);
    // E: class logits + regression outputs
    gemv4_kernel<H2N, false><<<(NCLS + 7) / 8, 256, 0, stream>>>(d3_w, nullptr,
                                                                 h2, cls, NCLS);
    gemv4_kernel<H2N, false><<<(NREG + 7) / 8, 256, 0, stream>>>(d4_w, nullptr,
                                                                 h2, out + 4 * NCLS,
                                                                 NREG);
    // F: softmax over classes
    softmax21_kernel<<<4, 32, 0, stream>>>(cls, out);
}